// MNR_15805479649896
// MI455X (gfx1250) — compile-verified
//
#include <hip/hip_runtime.h>
#include <math.h>
#include <stdint.h>
#include <utility>

typedef __attribute__((ext_vector_type(2))) float v2f;
typedef __attribute__((ext_vector_type(8))) float v8f;

#define DIM    256
#define NROWS  8192
#define CEPS   1e-8f
#define KC     32                  // K-floats staged per chunk (one 128B line/row)
#define NCHUNK (DIM / KC)          // 8
#define RS     (KC + 2)            // padded LDS row stride (floats): bank-conflict-free
#define ROWB   (RS * 4)            // LDS row stride bytes (136)
#define MATF   (16 * RS)           // floats per matrix chunk tile   (544)
#define MATB   (MATF * 4)          // bytes  (2176)
#define BUFF   (2 * MATF)          // floats per buffer (a+b)        (1088)
#define BUFB   (BUFF * 4)          // bytes  (4352)
#define WAVEF  (2 * BUFF)          // floats per wave (double buffer)(2176)
#define WAVES  4                   // waves per block -> 34.8 KB static LDS
#define GROWB  1024                // global row stride bytes

// One async copy = 32 lanes x 4B contiguous = exactly one 128B cacheline.
// ISA: the immediate hits
// BOTH sides, so IMM carries the LDS row placement and the SGPR base is
// pre-compensated by -IMM on the global side. All per-row variation is scalar.
template<int IMM>
__device__ __forceinline__ void async_cp(unsigned vdst, unsigned voff,
                                         unsigned long long sbase) {
    asm volatile("global_load_async_to_lds_b32 %0, %1, %2 offset:%3"
                 :: "v"(vdst), "v"(voff), "s"(sbase), "i"(IMM) : "memory");
}

template<int P, int M, int... Rs>
__device__ __forceinline__ void issue_rows_impl(std::integer_sequence<int, Rs...>,
                                                unsigned long long g0,
                                                unsigned vdst, unsigned voff) {
    // saddr(row r) = g0 + r*(1024 - 136); IMM(row r) = P*BUFB + M*MATB + r*136
    (async_cp<P * BUFB + M * MATB + Rs * ROWB>(
         vdst, voff, g0 + (unsigned long long)(Rs * (GROWB - ROWB))), ...);
}

template<int C>
__device__ __forceinline__ void issue_chunk(unsigned long long arow0,
                                            unsigned long long brow0,
                                            unsigned vdst, unsigned voff) {
    constexpr int P = C & 1;
    issue_rows_impl<P, 0>(std::make_integer_sequence<int, 16>{},
                          arow0 + C * (KC * 4) - (P * BUFB + 0 * MATB), vdst, voff);
    issue_rows_impl<P, 1>(std::make_integer_sequence<int, 16>{},
                          brow0 + C * (KC * 4) - (P * BUFB + 1 * MATB), vdst, voff);
}

template<int C>
__device__ __forceinline__ void chunk_step(const float* wbase, int r16, int half,
                                           unsigned long long arow0,
                                           unsigned long long brow0,
                                           unsigned vdst, unsigned voff,
                                           v8f& cab, v8f& caa, v8f& cbb) {
    // Async loads complete in order: <=32 outstanding means chunk C landed.
    if constexpr (C == NCHUNK - 1) asm volatile("s_wait_asynccnt 0"  ::: "memory");
    else                           asm volatile("s_wait_asynccnt 32" ::: "memory");

    constexpr int P = C & 1;
    const float* la = wbase + P * BUFF + r16 * RS + 2 * half;
    const float* lb = la + MATF;
#pragma unroll
    for (int kk = 0; kk < KC; kk += 4) {
        v2f fa = *(const v2f*)(la + kk);            // ds_load (2addr-fused), no conflicts
        v2f fb = *(const v2f*)(lb + kk);
        cab = __builtin_amdgcn_wmma_f32_16x16x4_f32(false, fa, false, fb,
                                                    (short)0, cab, false, false);
        caa = __builtin_amdgcn_wmma_f32_16x16x4_f32(false, fa, false, fa,
                                                    (short)0, caa, false, false);
        cbb = __builtin_amdgcn_wmma_f32_16x16x4_f32(false, fb, false, fb,
                                                    (short)0, cbb, false, false);
    }
    // s_wait_dscnt before the WMMAs retired the ds reads of buffer P, so
    // re-filling it with the next async chunk is safe.
    if constexpr (C + 2 < NCHUNK) issue_chunk<C + 2>(arow0, brow0, vdst, voff);
}

__global__ __launch_bounds__(WAVES * 32) void MNR_cosdiag_wmma_kernel(
    const float* __restrict__ a, const float* __restrict__ b,
    const int* __restrict__ label, float* __restrict__ row_loss)
{
    __shared__ float smem[WAVES * WAVEF];

    const int lane   = threadIdx.x & 31;
    // Wave id made provably uniform so all copy-address math stays in SGPRs.
    const int wave_u = __builtin_amdgcn_readfirstlane(threadIdx.x >> 5);
    const int tile   = blockIdx.x * WAVES + wave_u;   // 512 tiles of 16 rows
    const int row0   = tile * 16;
    const int half   = lane >> 4;                     // K half select (WMMA operand)
    const int r16    = lane & 15;                     // row/col within tile

    const float* wbase = &smem[wave_u * WAVEF];
    const unsigned lds0 = (unsigned)(size_t)wbase;    // wave's LDS byte offset

    // Loop-invariant per-lane VGPRs for every async copy in the kernel.
    const unsigned voff = (unsigned)(lane * 4);       // global 32-bit offset
    const unsigned vdst = lds0 + voff;                // LDS base + lane*4

    const unsigned long long arow0 =
        (unsigned long long)(const char*)a + (unsigned long long)row0 * GROWB;
    const unsigned long long brow0 =
        (unsigned long long)(const char*)b + (unsigned long long)row0 * GROWB;

    issue_chunk<0>(arow0, brow0, vdst, voff);
    issue_chunk<1>(arow0, brow0, vdst, voff);

    v8f cab = {};   // Gram(a,b) accumulator
    v8f caa = {};   // Gram(a,a)
    v8f cbb = {};   // Gram(b,b)

    chunk_step<0>(wbase, r16, half, arow0, brow0, vdst, voff, cab, caa, cbb);
    chunk_step<1>(wbase, r16, half, arow0, brow0, vdst, voff, cab, caa, cbb);
    chunk_step<2>(wbase, r16, half, arow0, brow0, vdst, voff, cab, caa, cbb);
    chunk_step<3>(wbase, r16, half, arow0, brow0, vdst, voff, cab, caa, cbb);
    chunk_step<4>(wbase, r16, half, arow0, brow0, vdst, voff, cab, caa, cbb);
    chunk_step<5>(wbase, r16, half, arow0, brow0, vdst, voff, cab, caa, cbb);
    chunk_step<6>(wbase, r16, half, arow0, brow0, vdst, voff, cab, caa, cbb);
    chunk_step<7>(wbase, r16, half, arow0, brow0, vdst, voff, cab, caa, cbb);

    // Diagonal (m,m): m in 0..7  -> VGPR m,   lane m
    //                 m in 8..15 -> VGPR m-8, lane m+16  (lanes 24..31)
    float dab = 0.f, daa = 0.f, dbb = 0.f;
#pragma unroll
    for (int r = 0; r < 8; ++r) {
        bool sel = (lane == r) || (lane == r + 24);
        if (sel) { dab = cab[r]; daa = caa[r]; dbb = cbb[r]; }
    }

    const bool is_diag = (lane < 8) || (lane >= 24);
    if (is_diag) {
        const int m  = (lane < 8) ? lane : (lane - 16);
        const int gr = row0 + m;
        const float na = fmaxf(sqrtf(daa), CEPS);   // torch eps clamp on norms
        const float nb = fmaxf(sqrtf(dbb), CEPS);
        const float cs = dab / (na * nb);           // cosine similarity, |cs|<=1
        const float z  = (float)label[gr];
        row_loss[gr] = log1pf(expf(cs)) - cs * z;   // softplus(x) - x*z
    }
}

// Deterministic fixed-order reduction of the 8192 per-row losses.
__global__ __launch_bounds__(256) void MNR_reduce_kernel(
    const float* __restrict__ row_loss, float* __restrict__ out)
{
    __shared__ float sm[256];
    const int t = threadIdx.x;
    float s = 0.f;
    for (int i = t; i < NROWS; i += 256) s += row_loss[i];
    sm[t] = s;
    __syncthreads();
#pragma unroll
    for (int off = 128; off > 0; off >>= 1) {
        if (t < off) sm[t] += sm[t + off];
        __syncthreads();
    }
    if (t == 0) out[0] = sm[0];
}

extern "C" void kernel_launch(void* const* d_in, const int* in_sizes, int n_in,
                              void* d_out, int out_size, void* d_ws, size_t ws_size,
                              hipStream_t stream) {
    const float* seq_a = (const float*)d_in[0];
    const float* seq_b = (const float*)d_in[1];
    const int*   label = (const int*)d_in[2];
    float* out      = (float*)d_out;
    float* row_loss = (float*)d_ws;                 // 8192 floats = 32 KB scratch

    // 512 tiles of 16 rows, 4 waves (wave32) per block -> 128 blocks
    MNR_cosdiag_wmma_kernel<<<128, WAVES * 32, 0, stream>>>(seq_a, seq_b, label, row_loss);
    MNR_reduce_kernel<<<1, 256, 0, stream>>>(row_loss, out);
}